// QLSTMLayer_76407468196162
// MI455X (gfx1250) — compile-verified
//
#include <hip/hip_runtime.h>
#include <math.h>

typedef __attribute__((ext_vector_type(2))) float v2f;
typedef __attribute__((ext_vector_type(8))) float v8f;

#define NQ        12
#define NST       4096     // 2^12 amplitudes
#define T_STEPS   32
#define H_DIM     128
#define IN_DIM    8
#define NTHREADS  256

__device__ __forceinline__ float2 cmulf(float2 a, float2 b) {
  return make_float2(a.x*b.x - a.y*b.y, a.x*b.y + a.y*b.x);
}
__device__ __forceinline__ float sigmoidf_(float x) { return 1.0f/(1.0f+__expf(-x)); }

// Composite basis permutation of CNOT ring 1 (r=1):  state_after[x] = state_before[rho(x)]
// sigma_w(x) = x ^ (bit_{11-w}(x) << (11-((w+1)%12))), applied w = 11 .. 0 (innermost first).
__device__ __forceinline__ unsigned rho_cnot1(unsigned x) {
  #pragma unroll
  for (int w = NQ-1; w >= 0; --w) {
    int pc = NQ-1-w;
    int pt = NQ-1-((w+1)%NQ);
    x ^= ((x >> pc) & 1u) << pt;
  }
  return x;
}

__global__ __launch_bounds__(NTHREADS) void qlstm_fused_kernel(
    const float* __restrict__ x,  const float* __restrict__ h0,
    const float* __restrict__ c0, const float* __restrict__ qw,
    const float* __restrict__ Wf, const float* __restrict__ bf,
    const float* __restrict__ Wi, const float* __restrict__ bi,
    const float* __restrict__ Wc, const float* __restrict__ bc,
    const float* __restrict__ Wo, const float* __restrict__ bo,
    float* __restrict__ out, int batch)
{
  __shared__ float2 st[NST];                       // 32 KB statevector (interleaved re,im)
  __shared__ __align__(16) float Mb[4][16][16];    // real 16x16 fused 3-qubit gates (layer 2)
  __shared__ float2 R0[NQ][4];                     // layer-1 Rot matrices (m00,m01,m10,m11)
  __shared__ float2 R1[NQ][4];                     // layer-2 Rot matrices
  __shared__ float2 aw[NQ][2];                     // per-wire state after RX+Rot1 on |0>
  __shared__ float2 prodHi[64], prodLo[64];        // partial-product tables for product state
  __shared__ unsigned short tabHi[64], tabLo[64];  // rho (CNOT ring 1) XOR tables
  __shared__ float hbuf[H_DIM], cbuf[H_DIM];
  __shared__ float qv[4];

  const int b    = blockIdx.x;
  const int tid  = threadIdx.x;
  const int lane = tid & 31;
  const int wv   = tid >> 5;     // wave id 0..7
  const int n    = lane & 15;
  const int half = lane >> 4;

  // --- measurement sign masks: conjugate Z_i through CNOT ring 2 (r=2) ---
  unsigned rows[NQ];
  #pragma unroll
  for (int p = 0; p < NQ; ++p) rows[p] = 1u << p;
  #pragma unroll
  for (int w = 0; w < NQ; ++w) {
    int pc = NQ-1-w, pt = NQ-1-((w+2)%NQ);
    rows[pt] ^= rows[pc];
  }
  const unsigned m0 = rows[11], m1 = rows[10], m2 = rows[9], m3 = rows[8];

  // --- one-time per-block setup ---
  if (tid < NQ) {
    #pragma unroll
    for (int l = 0; l < 2; ++l) {
      float phi = qw[(l*NQ + tid)*3 + 0];
      float th  = qw[(l*NQ + tid)*3 + 1];
      float om  = qw[(l*NQ + tid)*3 + 2];
      float cth = __cosf(0.5f*th), sth = __sinf(0.5f*th);
      float pp = 0.5f*(phi+om), qq = 0.5f*(phi-om);
      float2 ep = make_float2(__cosf(pp), -__sinf(pp));   // exp(-i(phi+om)/2)
      float2 em = make_float2(__cosf(qq), -__sinf(qq));   // exp(-i(phi-om)/2)
      float2* R = (l == 0) ? R0[tid] : R1[tid];
      R[0] = make_float2( ep.x*cth,  ep.y*cth);   // m00 = ep*c
      R[1] = make_float2(-em.x*sth,  em.y*sth);   // m01 = -conj(em)*s
      R[2] = make_float2( em.x*sth,  em.y*sth);   // m10 = em*s
      R[3] = make_float2( ep.x*cth, -ep.y*cth);   // m11 = conj(ep)*c
    }
  }
  if (tid < 64)        tabLo[tid]      = (unsigned short)rho_cnot1((unsigned)tid);
  else if (tid < 128)  tabHi[tid-64]   = (unsigned short)rho_cnot1((unsigned)(tid-64) << 6);
  if (tid < H_DIM) { hbuf[tid] = h0[b*H_DIM + tid]; cbuf[tid] = c0[b*H_DIM + tid]; }
  __syncthreads();

  // Build the four real 16x16 matrices M = real-embedding of kron(Rot2_{3j},Rot2_{3j+1},Rot2_{3j+2}).
  // Exactly 256 entries -> one per thread.
  {
    int j = tid >> 6, r = (tid >> 3) & 7, c2 = tid & 7;
    float2 u = cmulf(cmulf(R1[3*j  ][((r>>2)&1)*2 + ((c2>>2)&1)],
                           R1[3*j+1][((r>>1)&1)*2 + ((c2>>1)&1)]),
                           R1[3*j+2][( r     &1)*2 + ( c2    &1)]);
    Mb[j][2*r  ][2*c2  ] =  u.x;
    Mb[j][2*r  ][2*c2+1] = -u.y;
    Mb[j][2*r+1][2*c2  ] =  u.y;
    Mb[j][2*r+1][2*c2+1] =  u.x;
  }

  float wfr=0,bfr=0,wir=0,bir=0,wcr=0,bcr=0,wor=0,bor=0;
  if (tid < H_DIM) {
    wfr=Wf[tid]; bfr=bf[tid]; wir=Wi[tid]; bir=bi[tid];
    wcr=Wc[tid]; bcr=bc[tid]; wor=Wo[tid]; bor=bo[tid];
  }
  __syncthreads();

  for (int t = 0; t < T_STEPS; ++t) {
    // a_w = Rot1_w * RX(angle_w) |0>;  RX col0 = (cos(a/2), -i sin(a/2))
    if (tid < NQ) {
      float a  = (tid < IN_DIM) ? x[(b*T_STEPS + t)*IN_DIM + tid] : hbuf[tid - IN_DIM];
      float ca = __cosf(0.5f*a), sa = __sinf(0.5f*a);
      const float2* R = R0[tid];
      aw[tid][0] = make_float2(R[0].x*ca + R[1].y*sa, R[0].y*ca - R[1].x*sa);
      aw[tid][1] = make_float2(R[2].x*ca + R[3].y*sa, R[2].y*ca - R[3].x*sa);
    }
    if (tid >= 252) qv[tid - 252] = 0.0f;
    __syncthreads();

    if (tid < 64) {                       // wires 0..5 -> bits 11..6
      float2 p = aw[0][(tid>>5)&1];
      p = cmulf(p, aw[1][(tid>>4)&1]);
      p = cmulf(p, aw[2][(tid>>3)&1]);
      p = cmulf(p, aw[3][(tid>>2)&1]);
      p = cmulf(p, aw[4][(tid>>1)&1]);
      p = cmulf(p, aw[5][ tid    &1]);
      prodHi[tid] = p;
    } else if (tid < 128) {               // wires 6..11 -> bits 5..0
      int v = tid - 64;
      float2 p = aw[6][(v>>5)&1];
      p = cmulf(p, aw[7 ][(v>>4)&1]);
      p = cmulf(p, aw[8 ][(v>>3)&1]);
      p = cmulf(p, aw[9 ][(v>>2)&1]);
      p = cmulf(p, aw[10][(v>>1)&1]);
      p = cmulf(p, aw[11][ v    &1]);
      prodLo[v] = p;
    }
    __syncthreads();

    // Materialize psi after embedding + Rot layer 1 + CNOT ring 1 (permutation folded into read).
    #pragma unroll
    for (int k = 0; k < NST/NTHREADS; ++k) {
      unsigned xi = (unsigned)tid + NTHREADS*k;
      unsigned rx = (unsigned)tabHi[xi >> 6] ^ (unsigned)tabLo[xi & 63];
      st[xi] = cmulf(prodHi[rx >> 6], prodLo[rx & 63]);
    }
    __syncthreads();

    // Rot layer 2 as four fused 3-qubit gates: 16x16 real matrix x 512 groups via WMMA f32 16x16x4.
    #pragma unroll
    for (int j = 0; j < 4; ++j) {
      const int s = 9 - 3*j;              // low bit position of this triple's contiguous bit block
      v2f Bv[4];                          // B[k][n] = M[n][k]; lane n<16 holds K=0,1 ; n>=16 holds K=2,3
      #pragma unroll
      for (int kc = 0; kc < 4; ++kc)
        Bv[kc] = *(const v2f*)&Mb[j][n][4*kc + 2*half];
      #pragma unroll
      for (int tile = 0; tile < 4; ++tile) {
        const int tb = (wv + 8*tile) * 16;            // 16 groups per wave-tile
        const unsigned g   = (unsigned)(tb + n);
        const unsigned glo = g & ((1u << s) - 1u);
        const unsigned ghi = (g >> s) << (s + 3);
        v2f Av[4];                                    // A[m][k] = S[group m][comp k]
        #pragma unroll
        for (int kc = 0; kc < 4; ++kc) {
          unsigned amp = (unsigned)(2*kc + half);     // lane-half selects even/odd member
          unsigned idx = ghi | (amp << s) | glo;
          Av[kc] = *(const v2f*)&st[idx];             // ds_load_b64: one complex amplitude
        }
        v8f acc = {0.f,0.f,0.f,0.f,0.f,0.f,0.f,0.f};
        #pragma unroll
        for (int kc = 0; kc < 4; ++kc)
          acc = __builtin_amdgcn_wmma_f32_16x16x4_f32(false, Av[kc], false, Bv[kc],
                                                      (short)0, acc, false, false);
        const unsigned amp_o = (unsigned)(n >> 1);    // D[m][n] -> comp n of group tb+m
        const unsigned part  = (unsigned)(n & 1);
        #pragma unroll
        for (int r = 0; r < 8; ++r) {
          unsigned gg  = (unsigned)(tb + r + 8*half);
          unsigned idx = ((gg >> s) << (s + 3)) | (amp_o << s) | (gg & ((1u << s) - 1u));
          ((float*)&st[idx])[part] = acc[r];
        }
      }
      __syncthreads();
    }

    // <Z_i> with CNOT ring 2 absorbed into parity-sign masks.
    float p0=0.f, p1=0.f, p2=0.f, p3=0.f;
    #pragma unroll
    for (int k = 0; k < NST/NTHREADS; ++k) {
      unsigned xi = (unsigned)tid + NTHREADS*k;
      float2 cc = st[xi];
      float pr = cc.x*cc.x + cc.y*cc.y;
      p0 += (__popc(xi & m0) & 1) ? -pr : pr;
      p1 += (__popc(xi & m1) & 1) ? -pr : pr;
      p2 += (__popc(xi & m2) & 1) ? -pr : pr;
      p3 += (__popc(xi & m3) & 1) ? -pr : pr;
    }
    #pragma unroll
    for (int off = 16; off >= 1; off >>= 1) {
      p0 += __shfl_xor(p0, off, 32);
      p1 += __shfl_xor(p1, off, 32);
      p2 += __shfl_xor(p2, off, 32);
      p3 += __shfl_xor(p3, off, 32);
    }
    if (lane == 0) {
      atomicAdd(&qv[0], p0); atomicAdd(&qv[1], p1);
      atomicAdd(&qv[2], p2); atomicAdd(&qv[3], p3);
    }
    __syncthreads();

    // LSTM cell update (rank-1 gates, H=128).
    if (tid < H_DIM) {
      float fg = sigmoidf_(qv[0]*wfr + bfr);
      float ig = sigmoidf_(qv[1]*wir + bir);
      float ct = tanhf    (qv[2]*wcr + bcr);
      float og = sigmoidf_(qv[3]*wor + bor);
      float cn = fg*cbuf[tid] + ig*ct;
      cbuf[tid] = cn;
      hbuf[tid] = og * tanhf(cn);
    }
    __syncthreads();
  }

  if (tid < H_DIM) {
    out[b*H_DIM + tid]               = hbuf[tid];   // h_t
    out[batch*H_DIM + b*H_DIM + tid] = cbuf[tid];   // c_t
  }
}

extern "C" void kernel_launch(void* const* d_in, const int* in_sizes, int n_in,
                              void* d_out, int out_size, void* d_ws, size_t ws_size,
                              hipStream_t stream) {
  const float* x  = (const float*)d_in[0];
  const float* h0 = (const float*)d_in[1];
  const float* c0 = (const float*)d_in[2];
  const float* qw = (const float*)d_in[3];
  const float* Wf = (const float*)d_in[4];
  const float* bf = (const float*)d_in[5];
  const float* Wi = (const float*)d_in[6];
  const float* bi = (const float*)d_in[7];
  const float* Wc = (const float*)d_in[8];
  const float* bc = (const float*)d_in[9];
  const float* Wo = (const float*)d_in[10];
  const float* bo = (const float*)d_in[11];

  const int batch = in_sizes[1] / H_DIM;   // h0 is (B, H)

  qlstm_fused_kernel<<<batch, NTHREADS, 0, stream>>>(
      x, h0, c0, qw, Wf, bf, Wi, bi, Wc, bc, Wo, bo, (float*)d_out, batch);
}